// TemporalEdgesEmbedding_60644938219641
// MI455X (gfx1250) — compile-verified
//
#include <hip/hip_runtime.h>

// out = relu( reshape(x,(m,8)) @ W(8,360) + b ), out is (m,360) f32.
// Bandwidth-bound: 377MB store stream dominates (~16.5us floor @ 23.3TB/s).
// Compute path: V_WMMA_F32_16X16X4_F32, two per 16x16 tile (K=8).

typedef __attribute__((ext_vector_type(2)))  float v2f;
typedef __attribute__((ext_vector_type(8)))  float v8f;

#define NCOL  360
#define NPAD  368              // 23 * 16
#define NTILE 23               // tiles 0..21 fully in-range, tile 22 ragged
#define WAVES_PER_BLOCK 8
#define BLOCK (WAVES_PER_BLOCK * 32)

// One 16x16 output tile: C = bias + A(16x8) * B(8x16), via two K=4 WMMAs.
__device__ __forceinline__ v8f tile_mma(const float* sWt, const float* sB,
                                        int nt, int mloc, int half,
                                        v2f a1, v2f a2) {
    const int ncol = (nt << 4) + mloc;
    const float* wp = &sWt[ncol * 8 + half * 2];     // Wt[n][k] column-major
    const float bias = sB[ncol];
    v8f c = {bias, bias, bias, bias, bias, bias, bias, bias};
    v2f b1 = *(const v2f*)(wp);                      // K={2h, 2h+1}
    v2f b2 = *(const v2f*)(wp + 4);                  // K={4+2h, 5+2h}
    c = __builtin_amdgcn_wmma_f32_16x16x4_f32(
            false, a1, false, b1, (short)0, c, false, false);
    c = __builtin_amdgcn_wmma_f32_16x16x4_f32(
            false, a2, false, b2, (short)0, c, false, false);
    return c;
}

__device__ __forceinline__ float relu1(float v) {
    return __builtin_amdgcn_fmed3f(v, 0.0f, __builtin_inff());  // 1x v_med3_f32
}

__global__ __launch_bounds__(BLOCK) void fused_linear_relu_wmma(
    const float* __restrict__ x,   // m*8 floats (flat view of (N,2))
    const float* __restrict__ W,   // (8,360) row-major
    const float* __restrict__ b,   // (360,)
    float* __restrict__ out,       // (m,360)
    int m, long long nfloat)       // nfloat = total floats in x
{
    // LDS: W transposed to column-major Wt[n][k] (8 floats per output column),
    // zero-padded to 368 columns, plus padded bias. 13.2 KB total.
    __shared__ float sWt[NPAD * 8];
    __shared__ float sB[NPAD];

    const int tid = threadIdx.x;
    for (int i = tid; i < NPAD * 8; i += BLOCK) {
        int j = i >> 3, k = i & 7;
        sWt[i] = (j < NCOL) ? W[k * NCOL + j] : 0.0f;
    }
    for (int j = tid; j < NPAD; j += BLOCK)
        sB[j] = (j < NCOL) ? b[j] : 0.0f;
    __syncthreads();

    const int lane = tid & 31;
    const int wave = tid >> 5;
    const int half = lane >> 4;     // 0: lanes 0-15, 1: lanes 16-31
    const int mloc = lane & 15;     // row within M-tile / col within N-tile

    const long long chunkBase =
        ((long long)blockIdx.x * WAVES_PER_BLOCK + wave) * 16;
    if (chunkBase >= m) return;     // wave-uniform exit (after barrier)

    const long long row = chunkBase + mloc;

    // ---- A operand: 16x4 f32 layout. Lane holds row M=lane&15.
    // VGPR0/1 carry K={0,1} (lanes 0-15) or K={2,3} (lanes 16-31);
    // second WMMA uses K+4. Each lane: two 8B loads, reused for all 23 tiles.
    v2f a1 = {0.0f, 0.0f}, a2 = {0.0f, 0.0f};
    if (row < m) {
        const long long base = row * 8;
        if (base + 8 <= nfloat) {
            const v2f* xr = (const v2f*)(x + base);
            a1 = xr[half];          // floats {2h, 2h+1}
            a2 = xr[2 + half];      // floats {4+2h, 4+2h+1}
        } else {                    // ragged tail (zero-padded chunk)
            long long i0 = base + half * 2;
            a1.x = (i0     < nfloat) ? x[i0]     : 0.0f;
            a1.y = (i0 + 1 < nfloat) ? x[i0 + 1] : 0.0f;
            long long i1 = base + 4 + half * 2;
            a2.x = (i1     < nfloat) ? x[i1]     : 0.0f;
            a2.y = (i1 + 1 < nfloat) ? x[i1 + 1] : 0.0f;
        }
    }

    // Per-lane output pointer for D: VGPR v -> row chunkBase + 8*half + v,
    // col = nt*16 + mloc. Advance by 64B per tile.
    float* po = out + (chunkBase + half * 8) * (long long)NCOL + mloc;

    if (chunkBase + 16 <= (long long)m) {
        // ---- Common case: all 16 rows in range.
        // Tiles 0..21: ncol <= 351 < 360 for every lane -> zero predication:
        // ds loads, 2 wmma, 8 med3, s_clause + 8 plain global_store_b32
        // (each two contiguous 64B segments, merged in the 192MB L2).
        #pragma unroll 1
        for (int nt = 0; nt < NTILE - 1; ++nt) {
            v8f c = tile_mma(sWt, sB, nt, mloc, half, a1, a2);
            #pragma unroll
            for (int v = 0; v < 8; ++v)
                po[v * NCOL] = relu1(c[v]);
            po += 16;
        }
        // Tile 22: lanes with mloc >= 8 are past column 359 (one v_cmpx).
        {
            v8f c = tile_mma(sWt, sB, NTILE - 1, mloc, half, a1, a2);
            if (((NTILE - 1) << 4) + mloc < NCOL) {
                #pragma unroll
                for (int v = 0; v < 8; ++v)
                    po[v * NCOL] = relu1(c[v]);
            }
        }
    } else {
        // ---- At most one ragged-M wave in the entire grid.
        for (int nt = 0; nt < NTILE; ++nt) {
            v8f c = tile_mma(sWt, sB, nt, mloc, half, a1, a2);
            const int ncol = (nt << 4) + mloc;
            if (ncol < NCOL) {
                const long long rbase = chunkBase + half * 8;
                #pragma unroll
                for (int v = 0; v < 8; ++v) {
                    const long long r = rbase + v;
                    if (r < m)
                        out[r * NCOL + ncol] = relu1(c[v]);
                }
            }
        }
    }
}

extern "C" void kernel_launch(void* const* d_in, const int* in_sizes, int n_in,
                              void* d_out, int out_size, void* d_ws, size_t ws_size,
                              hipStream_t stream) {
    const float* x = (const float*)d_in[0];
    const float* W = (const float*)d_in[1];
    const float* b = (const float*)d_in[2];
    float* out = (float*)d_out;

    const long long nfloat = (long long)in_sizes[0];   // N_ROWS * 2
    const long long nrows  = nfloat / 2;
    const int m = (int)((nrows + 3) / 4);              // number of 8-wide chunks

    const int waves  = (m + 15) / 16;
    int blocks = (waves + WAVES_PER_BLOCK - 1) / WAVES_PER_BLOCK;
    if (blocks < 1) blocks = 1;

    fused_linear_relu_wmma<<<blocks, BLOCK, 0, stream>>>(x, W, b, out, m, nfloat);
}